// SparseFMLayer_71416716198136
// MI455X (gfx1250) — compile-verified
//
#include <hip/hip_runtime.h>
#include <hip/hip_bf16.h>

typedef float v2f __attribute__((ext_vector_type(2)));
typedef float v8f __attribute__((ext_vector_type(8)));

#define FM_K 16

// ---------------------------------------------------------------------------
// Pass 0: sentinel-init the row-range arrays (must happen every launch; the
// harness does not re-poison between graph replays).
// ---------------------------------------------------------------------------
__global__ void fm_init_rowptr(int* __restrict__ rstart, int* __restrict__ rend, int B) {
  int b = blockIdx.x * blockDim.x + threadIdx.x;
  if (b < B) { rstart[b] = 0x7FFFFFFF; rend[b] = 0; }
}

// ---------------------------------------------------------------------------
// Pass 1: rows[] is sorted, so each segment is a contiguous [start,end) range.
// Recover it with atomicMin/atomicMax (2 int atomics per nnz, L2-resident).
// ---------------------------------------------------------------------------
__global__ void fm_rowptr(const int* __restrict__ rows, int nnz,
                          int* __restrict__ rstart, int* __restrict__ rend) {
  int i = blockIdx.x * blockDim.x + threadIdx.x;
  if (i < nnz) {
    int r = rows[i];
    atomicMin(&rstart[r], i);
    atomicMax(&rend[r], i + 1);
  }
}

// ---------------------------------------------------------------------------
// Pass 2: one wave32 per output row. Lane l owns k = l&15; half h = l>>4
// processes entries start+h, start+h+2, ... Embedding row reads are 16
// consecutive floats (64B) per entry -> coalesced across lanes 0-15 / 16-31.
// All accumulation is in registers; cross-half combine via shfl_xor.
// Writes xv[b, 0..15], lin[b], sq[b] = sum_k sum_e (v*ev)^2 unconditionally
// (so no zero-init of those buffers is needed).
// ---------------------------------------------------------------------------
__global__ void fm_rows(const float* __restrict__ vals,
                        const int*   __restrict__ cols,
                        const float* __restrict__ weight,
                        const float* __restrict__ emb,
                        const int*   __restrict__ rstart,
                        const int*   __restrict__ rend,
                        float* __restrict__ xv,
                        float* __restrict__ lin,
                        float* __restrict__ sq,
                        int B) {
  int lane = threadIdx.x & 31;
  int row  = (blockIdx.x * blockDim.x + threadIdx.x) >> 5;
  if (row >= B) return;

  int s = rstart[row];
  int e = rend[row];
  if (s > e) { s = 0; e = 0; }          // empty segment (sentinel untouched)

  const int k = lane & 15;
  const int h = lane >> 4;

  float acc  = 0.0f;   // xv contribution for this (k, parity)
  float asq  = 0.0f;   // sum of (v*ev)^2 over this lane's subset
  float alin = 0.0f;   // linear term (only k==0 lanes accumulate)

  for (int i = s + h; i < e; i += 2) {
    float v  = vals[i];
    int   c  = cols[i];
    float ev = emb[(size_t)c * FM_K + k];
    float t  = v * ev;
    acc += t;
    asq += t * t;                        // v^2 * ev^2 == (v*ev)^2
    if (k == 0) alin += v * weight[c];
  }

  // combine even/odd halves for the per-k xv and the linear term
  acc  += __shfl_xor(acc, 16, 32);
  alin += __shfl_xor(alin, 16, 32);
  // full 32-lane butterfly for the scalar sum-of-squares
  #pragma unroll
  for (int off = 16; off >= 1; off >>= 1) asq += __shfl_xor(asq, off, 32);

  if (lane < FM_K) xv[(size_t)row * FM_K + k] = acc;
  if (lane == 0) { lin[row] = alin; sq[row] = asq; }
}

// ---------------------------------------------------------------------------
// Pass 3: per 16-row tile of xv [16 x 16], compute G = X @ X^T with four
// chained V_WMMA_F32_16X16X4_F32 (exact f32, RNE). Per ISA 7.12.2, the
// 16x4 f32 A operand and 4x16 f32 B operand have the SAME per-lane layout:
// lane l holds row (l%16), K = 4*kb + 2*(l/16) + {0,1}. So one float2 pair
// feeds both A and B. diag(G)[m] = sum_k xv[m,k]^2. Diagonal element (m,m)
// sits in C-VGPR m%8 at lane m (m<8) or lane m+16 (m>=8).
// out[b] = bias + lin[b] + 0.5*(diag - sq[b]).
// ---------------------------------------------------------------------------
__global__ void fm_finalize(const float* __restrict__ xv,
                            const float* __restrict__ lin,
                            const float* __restrict__ sq,
                            const float* __restrict__ bias,
                            float* __restrict__ out,
                            int B) {
  int lane = threadIdx.x & 31;
  int tile = (blockIdx.x * blockDim.x + threadIdx.x) >> 5;
  int tiles = B >> 4;                    // B is a multiple of 16
  if (tile >= tiles) return;             // wave-uniform branch; EXEC stays all-1s

  const int m = lane & 15;
  const int h = lane >> 4;
  const float* rowp = xv + ((size_t)tile * 16 + m) * FM_K + 2 * h;

  v2f a0 = *(const v2f*)(rowp + 0);      // K = 4*0 + 2h + {0,1}
  v2f a1 = *(const v2f*)(rowp + 4);      // K = 4*1 + 2h + {0,1}
  v2f a2 = *(const v2f*)(rowp + 8);
  v2f a3 = *(const v2f*)(rowp + 12);

  v8f c = {};
  c = __builtin_amdgcn_wmma_f32_16x16x4_f32(false, a0, false, a0, (short)0, c, false, false);
  c = __builtin_amdgcn_wmma_f32_16x16x4_f32(false, a1, false, a1, (short)0, c, false, false);
  c = __builtin_amdgcn_wmma_f32_16x16x4_f32(false, a2, false, a2, (short)0, c, false, false);
  c = __builtin_amdgcn_wmma_f32_16x16x4_f32(false, a3, false, a3, (short)0, c, false, false);

  // Diagonal extraction, branch-free: lanes 0-7 hold (m=lane) in c[lane];
  // lanes 24-31 hold (m=lane-16) in c[lane-24]; other lanes idle.
  // Masked accumulation lowers to flat v_cmp/v_cndmask/v_add (no EXEC churn).
  const bool active = (lane < 8) | (lane >= 24);
  const int  idx    = (lane < 8) ? lane : (lane - 24);   // 0..7 where active
  const int  dm     = (lane < 8) ? lane : (lane - 16);   // diag row in tile

  float diag = 0.0f;
  #pragma unroll
  for (int i = 0; i < 8; ++i) diag += (idx == i) ? c[i] : 0.0f;

  if (active) {
    int b = tile * 16 + dm;
    out[b] = bias[0] + lin[b] + 0.5f * (diag - sq[b]);
  }
}

// ---------------------------------------------------------------------------
// Launch: inputs are (vals f32[NNZ], rows i32[NNZ], cols i32[NNZ],
// weight f32[D], embedding f32[D*16], bias f32[1], batch i32[1]).
// Output: f32[B]. Workspace layout (floats/ints), total 320 KB @ B=4096:
//   xv[B*16] | lin[B] | sq[B] | rstart[B] | rend[B]
// ---------------------------------------------------------------------------
extern "C" void kernel_launch(void* const* d_in, const int* in_sizes, int n_in,
                              void* d_out, int out_size, void* d_ws, size_t ws_size,
                              hipStream_t stream) {
  const float* vals   = (const float*)d_in[0];
  const int*   rows   = (const int*)  d_in[1];
  const int*   cols   = (const int*)  d_in[2];
  const float* weight = (const float*)d_in[3];
  const float* emb    = (const float*)d_in[4];
  const float* bias   = (const float*)d_in[5];

  const int nnz = in_sizes[0];
  const int B   = out_size;                  // [B,1] output, flat B floats
  float* out = (float*)d_out;

  float* xv     = (float*)d_ws;              // B*16 floats
  float* lin    = xv + (size_t)B * FM_K;     // B floats
  float* sq     = lin + B;                   // B floats
  int*   rstart = (int*)(sq + B);            // B ints
  int*   rend   = rstart + B;                // B ints

  fm_init_rowptr<<<(B + 255) / 256, 256, 0, stream>>>(rstart, rend, B);
  fm_rowptr<<<(nnz + 255) / 256, 256, 0, stream>>>(rows, nnz, rstart, rend);

  // one wave32 per row
  int rows_threads = B * 32;
  fm_rows<<<(rows_threads + 255) / 256, 256, 0, stream>>>(
      vals, cols, weight, emb, rstart, rend, xv, lin, sq, B);

  // one wave32 per 16-row tile
  int tiles = B / 16;
  int fin_threads = tiles * 32;
  fm_finalize<<<(fin_threads + 255) / 256, 256, 0, stream>>>(
      xv, lin, sq, bias, out, B);
}